// SimpleBlock1d_kinet2_47682726920392
// MI455X (gfx1250) — compile-verified
//
#include <hip/hip_runtime.h>
#include <math.h>

// ---------------- problem constants ----------------
#define BS    2
#define SN    512
#define WD    64
#define NMODE 16
#define PPAIR 130816                 // S*(S-1)/2
#define HSZ   (BS*WD*SN)             // 65536 floats
#define GSZ   (BS*SN*SN)             // 524288 floats
#define INV_SQRT_N    0.04419417382415922f   // 1/sqrt(512)
#define TWO_PI_OVER_N 0.01227184630308513f   // 2*pi/512

typedef float v2f __attribute__((ext_vector_type(2)));
typedef float v8f __attribute__((ext_vector_type(8)));

// ---------------- fp32 WMMA fragment helpers (V_WMMA_F32_16X16X4_F32) ----------------
// Layout per CDNA5 ISA 7.12.2: lane%16 = M/N index, lane/16 selects K+2,
// VGPR r holds K = k0 + r + 2*(lane/16).
__device__ __forceinline__ v2f frag_a(const float* __restrict__ A, int am, int ak,
                                      int ka, int mn) {
  v2f av;
  av[0] = A[mn*am + ka*ak];
  av[1] = A[mn*am + (ka+1)*ak];
  return av;
}
__device__ __forceinline__ v2f frag_b(const float* __restrict__ B, int bk, int bn,
                                      int ka, int mn) {
  v2f bv;
  bv[0] = B[ka*bk + mn*bn];
  bv[1] = B[(ka+1)*bk + mn*bn];
  return bv;
}
__device__ __forceinline__ v8f wmma4(v2f a, v2f b, v8f c) {
  return __builtin_amdgcn_wmma_f32_16x16x4_f32(false, a, false, b, (short)0, c,
                                               false, false);
}
__device__ __forceinline__ void wmma_store_f32(float* __restrict__ D, int dm, int dn, v8f acc) {
  const int lane = threadIdx.x & 31;
  const int n  = lane & 15;
  const int hf = lane >> 4;
#pragma unroll
  for (int r = 0; r < 8; ++r) D[(r + 8*hf)*dm + n*dn] = acc[r];
}

// ---------------- elementwise / small kernels ----------------
__global__ void k_fc0(const float* __restrict__ x, const float* __restrict__ w,
                      const float* __restrict__ bias, float* __restrict__ h) {
  int t = blockIdx.x*blockDim.x + threadIdx.x;               // HSZ
  int b = t >> 15, rem = t & 32767, c = rem >> 9, n = rem & 511;
  h[t] = x[b*1024 + n*2 + 0]*w[c] + x[b*1024 + n*2 + 1]*w[WD + c] + bias[c];
}

// rfft (ortho) restricted to first 16 modes: direct DFT
__global__ void k_dft(const float* __restrict__ h, float* __restrict__ Xr, float* __restrict__ Xi) {
  int t = blockIdx.x*blockDim.x + threadIdx.x;               // BS*WD*NMODE = 2048
  int b = t >> 10, rem = t & 1023, c = rem >> 4, m = rem & 15;
  const float* hp = h + b*(WD*SN) + c*SN;
  float w = TWO_PI_OVER_N * (float)m;
  float fr = 0.f, fi = 0.f;
  for (int n = 0; n < SN; ++n) {
    float s, co; sincosf(w*(float)n, &s, &co);
    fr += hp[n]*co; fi -= hp[n]*s;
  }
  Xr[t] = fr * INV_SQRT_N; Xi[t] = fi * INV_SQRT_N;
}

// out_m[b,o,m] = sum_i X[b,i,m] * (wr + j wi)[i,o,m]
__global__ void k_einsum(const float* __restrict__ Xr, const float* __restrict__ Xi,
                         const float* __restrict__ sw, float* __restrict__ Yr,
                         float* __restrict__ Yi) {
  int t = blockIdx.x*blockDim.x + threadIdx.x;               // 2048
  int b = t >> 10, rem = t & 1023, o = rem >> 4, m = rem & 15;
  float yr = 0.f, yi = 0.f;
  for (int i = 0; i < WD; ++i) {
    float xr = Xr[b*1024 + i*16 + m];
    float xi = Xi[b*1024 + i*16 + m];
    const float* wp = sw + ((size_t)(i*WD + o)*NMODE + m)*2;
    float wr = wp[0], wi = wp[1];
    yr += xr*wr - xi*wi;
    yi += xr*wi + xi*wr;
  }
  Yr[t] = yr; Yi[t] = yi;
}

// irfft (ortho): a[n] = (Re Y0 + 2*sum_{m=1..15} (Yr cos - Yi sin)) / sqrt(N)
__global__ void k_idft(const float* __restrict__ Yr, const float* __restrict__ Yi,
                       float* __restrict__ a) {
  int t = blockIdx.x*blockDim.x + threadIdx.x;               // HSZ
  int b = t >> 15, rem = t & 32767, o = rem >> 9, n = rem & 511;
  const float* yr = Yr + b*1024 + o*16;
  const float* yi = Yi + b*1024 + o*16;
  float acc = yr[0];
  for (int m = 1; m < NMODE; ++m) {
    float s, co; sincosf(TWO_PI_OVER_N*(float)(m*n), &s, &co);
    acc += 2.f*(yr[m]*co - yi[m]*s);
  }
  a[t] = acc * INV_SQRT_N;
}

// ---------------- WMMA matmul kernels ----------------
// p[b,o,n] = sum_i Wc[o,i] h[b,i,n] + bc[o].
// One wave computes the full 64-row column block (4 M-tiles) for one 16-col
// strip: B fragment loaded once per K-step, reused by 4 WMMAs.
__global__ void k_conv(const float* __restrict__ hbuf, const float* __restrict__ Wc,
                       const float* __restrict__ bc, float* __restrict__ p) {
  int wid = (blockIdx.x*blockDim.x + threadIdx.x) >> 5;      // 64 waves total
  int b = wid >> 5, nt = wid & 31;
  const int lane = threadIdx.x & 31;
  const int mn = lane & 15, hf = lane >> 4;
  const float* hb = hbuf + b*(WD*SN) + nt*16;
  v8f acc[4] = {};
  for (int k0 = 0; k0 < WD; k0 += 4) {
    const int ka = k0 + 2*hf;
    v2f bv = frag_b(hb, SN, 1, ka, mn);
#pragma unroll
    for (int mt = 0; mt < 4; ++mt) {
      v2f av = frag_a(Wc + mt*16*WD, WD, 1, ka, mn);
      acc[mt] = wmma4(av, bv, acc[mt]);
    }
  }
#pragma unroll
  for (int mt = 0; mt < 4; ++mt) {
#pragma unroll
    for (int r = 0; r < 8; ++r) {
      int row = mt*16 + r + 8*hf;
      p[b*(WD*SN) + row*SN + nt*16 + mn] = acc[mt][r] + bc[row];
    }
  }
}

// Gram matrix G[b,i,j] = sum_c src[b,c,i]*src[b,c,j]  (M=N=512, K=64).
// One wave computes a 32x32 region (2x2 tiles): 4 WMMAs per 2 A + 2 B fragments.
__global__ void k_gram(const float* __restrict__ src, float* __restrict__ G) {
  int wid = (blockIdx.x*blockDim.x + threadIdx.x) >> 5;      // 512 waves
  int b = wid >> 8, rem = wid & 255, it2 = rem >> 4, jt2 = rem & 15;
  const int i0 = it2*32, j0 = jt2*32;
  const int lane = threadIdx.x & 31;
  const int mn = lane & 15, hf = lane >> 4;
  const float* sb = src + b*(WD*SN);
  v8f a00 = {}, a01 = {}, a10 = {}, a11 = {};
  for (int k0 = 0; k0 < WD; k0 += 4) {
    const int ka = k0 + 2*hf;
    v2f av0 = frag_a(sb + i0,      1, SN, ka, mn);   // A[m][k] = src[b,k,i0+m]
    v2f av1 = frag_a(sb + i0 + 16, 1, SN, ka, mn);
    v2f bv0 = frag_b(sb + j0,      SN, 1, ka, mn);   // B[k][n] = src[b,k,j0+n]
    v2f bv1 = frag_b(sb + j0 + 16, SN, 1, ka, mn);
    a00 = wmma4(av0, bv0, a00);
    a01 = wmma4(av0, bv1, a01);
    a10 = wmma4(av1, bv0, a10);
    a11 = wmma4(av1, bv1, a11);
  }
  float* gb = G + b*(SN*SN);
  wmma_store_f32(gb + (i0     )*SN + j0,      SN, 1, a00);
  wmma_store_f32(gb + (i0     )*SN + j0 + 16, SN, 1, a01);
  wmma_store_f32(gb + (i0 + 16)*SN + j0,      SN, 1, a10);
  wmma_store_f32(gb + (i0 + 16)*SN + j0 + 16, SN, 1, a11);
}

// T[b,c,i] = sum_j V[b,c,j]*cm[b,i,j]  (cm symmetric -> B[k][n] = cm[b,k,i0+n]).
// K=512; one wave does all 4 c-tiles for one i-strip: B fragment reused 4x.
__global__ void k_Tmm(const float* __restrict__ V, const float* __restrict__ cm,
                      float* __restrict__ T) {
  int wid = (blockIdx.x*blockDim.x + threadIdx.x) >> 5;      // 64 waves
  int b = wid >> 5, it = wid & 31;
  const int lane = threadIdx.x & 31;
  const int mn = lane & 15, hf = lane >> 4;
  const float* vb = V + b*(WD*SN);
  const float* cb = cm + b*(SN*SN) + it*16;
  v8f acc[4] = {};
  for (int k0 = 0; k0 < SN; k0 += 4) {
    const int ka = k0 + 2*hf;
    v2f bv = frag_b(cb, SN, 1, ka, mn);
#pragma unroll
    for (int ct = 0; ct < 4; ++ct) {
      v2f av = frag_a(vb + ct*16*SN, SN, 1, ka, mn);
      acc[ct] = wmma4(av, bv, acc[ct]);
    }
  }
#pragma unroll
  for (int ct = 0; ct < 4; ++ct)
    wmma_store_f32(T + b*(WD*SN) + ct*16*SN + it*16, SN, 1, acc[ct]);
}

// ---------------- DSMC pieces ----------------
__global__ void k_norms(const float* __restrict__ src, float* __restrict__ dst) {
  int t = blockIdx.x*blockDim.x + threadIdx.x;               // BS*SN
  int b = t >> 9, n = t & 511;
  float s = 0.f;
  for (int c = 0; c < WD; ++c) { float v = src[b*(WD*SN) + c*SN + n]; s += v*v; }
  dst[t] = s;
}

__global__ void k_vmax0(float* vmax) { if (threadIdx.x < BS) vmax[threadIdx.x] = 0.f; }

// Gram -> distances in place; per-batch max(v_r) via wave-reduce + atomic on float bits
__global__ void k_rdist(float* __restrict__ gx, float* __restrict__ gv,
                        const float* __restrict__ np_, const float* __restrict__ na_,
                        float* __restrict__ vmax) {
  int t = blockIdx.x*blockDim.x + threadIdx.x;               // GSZ
  int b = t >> 18, rem = t & 262143, i = rem >> 9, j = rem & 511;
  float xr = sqrtf(fmaxf(np_[b*SN + i] + np_[b*SN + j] - 2.f*gx[t], 0.f));
  float vr = sqrtf(fmaxf(na_[b*SN + i] + na_[b*SN + j] - 2.f*gv[t], 0.f));
  gx[t] = xr; gv[t] = vr;
  float m = vr;
  for (int d = 16; d; d >>= 1) m = fmaxf(m, __shfl_xor(m, d, 32));
  if ((threadIdx.x & 31) == 0)
    atomicMax((unsigned int*)(vmax + b), __float_as_uint(m));  // vr >= 0: bit order == float order
}

__global__ void k_cm(const float* __restrict__ xr, const float* __restrict__ vr,
                     const float* __restrict__ vmax, float* __restrict__ cm) {
  int t = blockIdx.x*blockDim.x + threadIdx.x;               // GSZ
  int b = t >> 18;
  float vm = fmaxf(vmax[b], 1e-30f);
  float mask = vr[t]/vm * expf(-xr[t]);
  cm[t] = (mask > 0.1f) ? 1.f : 0.f;
}

__global__ void k_rowsum(const float* __restrict__ cm, float* __restrict__ Rv) {
  int t = blockIdx.x*blockDim.x + threadIdx.x;               // BS*SN
  int b = t >> 9, i = t & 511;
  const float* row = cm + b*(SN*SN) + i*SN;
  float s = 0.f;
  for (int j = 0; j < SN; ++j) s += row[j];
  Rv[t] = s;
}

// W[b,c,i] = sum_p cm[i,p] * sgn(i-p) * v_r[i,p] * rvec[b,c,pidx(min,max)]
// (partner index p < output index i  =>  +proj; p > i => -proj)
__global__ void k_W(const float* __restrict__ cm, const float* __restrict__ vr,
                    const float* __restrict__ rvec, float* __restrict__ Wout) {
  __shared__ float coeff[SN];
  __shared__ int   pidx[SN];
  int blk = blockIdx.x;                                      // BS*SN
  int b = blk >> 9, i = blk & 511;
  int t = threadIdx.x;                                       // 256
  for (int j = t; j < SN; j += 256) {
    float c = cm[b*(SN*SN) + i*SN + j];
    float cf = 0.f; int pi = 0;
    if (c != 0.f && j != i) {
      int lo = (j < i) ? j : i, hi = (j < i) ? i : j;
      pi = lo*SN - (lo*(lo+1))/2 + hi - lo - 1;
      float sgn = (j < i) ? 1.f : -1.f;
      cf = sgn * vr[b*(SN*SN) + i*SN + j];
    }
    coeff[j] = cf; pidx[j] = pi;
  }
  __syncthreads();
  int wave = t >> 5, lane = t & 31;
  for (int c = wave; c < WD; c += 8) {
    const float* rp = rvec + (size_t)(b*WD + c)*PPAIR;
    float acc = 0.f;
    for (int j = lane; j < SN; j += 32) {
      if (j + 32 < SN) __builtin_prefetch(&rp[pidx[j + 32]], 0, 1);  // global_prefetch_b8
      float cf = coeff[j];
      if (cf != 0.f) acc += cf * rp[pidx[j]];
    }
    for (int d = 16; d; d >>= 1) acc += __shfl_xor(acc, d, 32);
    if (lane == 0) Wout[b*(WD*SN) + c*SN + i] = acc;
  }
}

// v_new = v*(1+0.5*Rv) - 0.5*Tv + W ; x_new = 0.5*(x*(R+1)+Tx)/R + v_new ; R=Rv+1
__global__ void k_update(const float* __restrict__ p, const float* __restrict__ a,
                         const float* __restrict__ Tx, const float* __restrict__ Tv,
                         const float* __restrict__ W, const float* __restrict__ Rv,
                         float* __restrict__ h, int do_relu) {
  int t = blockIdx.x*blockDim.x + threadIdx.x;               // HSZ
  int b = t >> 15, rem = t & 32767, i = rem & 511;
  float vi = a[t], xi = p[t];
  float rv = Rv[b*SN + i];
  float vn = vi*(1.f + 0.5f*rv) - 0.5f*Tv[t] + W[t];
  float R  = rv + 1.f;
  float xn = 0.5f*(xi*(R + 1.f) + Tx[t])/R + vn;
  h[t] = (do_relu && xn < 0.f) ? 0.f : xn;
}

// fc1(relu) + fc2: block per (b,s), 128 threads = hidden units
__global__ void k_head(const float* __restrict__ h, const float* __restrict__ w1,
                       const float* __restrict__ b1, const float* __restrict__ w2,
                       const float* __restrict__ b2, float* __restrict__ out) {
  __shared__ float red[128];
  int blk = blockIdx.x;                                      // BS*SN
  int b = blk >> 9, s = blk & 511;
  int k = threadIdx.x;                                       // 128
  float acc = b1[k];
  for (int c = 0; c < WD; ++c) acc += h[b*(WD*SN) + c*SN + s] * w1[c*128 + k];
  acc = fmaxf(acc, 0.f) * w2[k];
  red[k] = acc; __syncthreads();
  for (int d = 64; d; d >>= 1) { if (k < d) red[k] += red[k + d]; __syncthreads(); }
  if (k == 0) out[blk] = red[0] + b2[0];
}

// ---------------- host launcher ----------------
extern "C" void kernel_launch(void* const* d_in, const int* in_sizes, int n_in,
                              void* d_out, int out_size, void* d_ws, size_t ws_size,
                              hipStream_t stream) {
  (void)in_sizes; (void)n_in; (void)out_size; (void)ws_size;
  const float* x      = (const float*)d_in[0];
  // d_in[1] = v : unused by the reference math (overwritten by a inside dsmc)
  const float* vecs   = (const float*)d_in[2];
  const float* fc0_w  = (const float*)d_in[3];
  const float* fc0_b  = (const float*)d_in[4];
  const float* spec_w = (const float*)d_in[5];
  const float* conv_w = (const float*)d_in[6];
  const float* conv_b = (const float*)d_in[7];
  const float* fc1_w  = (const float*)d_in[8];
  const float* fc1_b  = (const float*)d_in[9];
  const float* fc2_w  = (const float*)d_in[10];
  const float* fc2_b  = (const float*)d_in[11];
  float* out = (float*)d_out;

  // workspace layout (~7.9 MB of f32)
  float* ws   = (float*)d_ws;
  float* h    = ws;                  // HSZ
  float* a    = h   + HSZ;           // HSZ
  float* p    = a   + HSZ;           // HSZ
  float* Tx   = p   + HSZ;           // HSZ
  float* Tv   = Tx  + HSZ;           // HSZ
  float* W    = Tv  + HSZ;           // HSZ
  float* Gx   = W   + HSZ;           // GSZ  (Gram of p -> x_r in place)
  float* Gv   = Gx  + GSZ;           // GSZ  (Gram of a -> v_r in place)
  float* cm   = Gv  + GSZ;           // GSZ
  float* np_  = cm  + GSZ;           // BS*SN
  float* na_  = np_ + BS*SN;         // BS*SN
  float* Rv   = na_ + BS*SN;         // BS*SN
  float* vmax = Rv  + BS*SN;         // BS (padded to 32)
  float* Xr   = vmax + 32;           // BS*WD*NMODE = 2048
  float* Xi   = Xr + 2048;
  float* Yr   = Xi + 2048;
  float* Yi   = Yr + 2048;

  k_fc0<<<HSZ/256, 256, 0, stream>>>(x, fc0_w, fc0_b, h);

  for (int L = 0; L < 4; ++L) {
    // spectral conv -> a
    k_dft   <<<2048/256, 256, 0, stream>>>(h, Xr, Xi);
    k_einsum<<<2048/256, 256, 0, stream>>>(Xr, Xi, spec_w + (size_t)L*WD*WD*NMODE*2, Yr, Yi);
    k_idft  <<<HSZ/256, 256, 0, stream>>>(Yr, Yi, a);
    // pointwise conv -> p (WMMA, 4 tiles/wave)
    k_conv  <<<16, 128, 0, stream>>>(h, conv_w + (size_t)L*WD*WD, conv_b + (size_t)L*WD, p);
    // pairwise distances via Gram (WMMA, 2x2 tiles/wave)
    k_gram  <<<128, 128, 0, stream>>>(p, Gx);
    k_gram  <<<128, 128, 0, stream>>>(a, Gv);
    k_norms <<<4, 256, 0, stream>>>(p, np_);
    k_norms <<<4, 256, 0, stream>>>(a, na_);
    k_vmax0 <<<1, 32, 0, stream>>>(vmax);
    k_rdist <<<GSZ/256, 256, 0, stream>>>(Gx, Gv, np_, na_, vmax);
    k_cm    <<<GSZ/256, 256, 0, stream>>>(Gx, Gv, vmax, cm);
    k_rowsum<<<4, 256, 0, stream>>>(cm, Rv);
    // partner sums against mask matrix (WMMA, K=512, 4 tiles/wave)
    k_Tmm   <<<16, 128, 0, stream>>>(p, cm, Tx);
    k_Tmm   <<<16, 128, 0, stream>>>(a, cm, Tv);
    // random-unit-vector antisymmetric term
    k_W     <<<BS*SN, 256, 0, stream>>>(cm, Gv, vecs + (size_t)L*BS*WD*PPAIR, W);
    // DSMC update -> next h (relu on layers 0..2)
    k_update<<<HSZ/256, 256, 0, stream>>>(p, a, Tx, Tv, W, Rv, h, (L < 3) ? 1 : 0);
  }

  k_head<<<BS*SN, 128, 0, stream>>>(h, fc1_w, fc1_b, fc2_w, fc2_b, out);
}